// PostProcessor3D_75651553952297
// MI455X (gfx1250) — compile-verified
//
#include <hip/hip_runtime.h>
#include <cstdint>

#define USE_ASYNC_LDS 1

namespace {
constexpr int   Dz     = 64;
constexpr int   Hy     = 512;
constexpr int   Wx     = 512;
constexpr int   RAD    = 2;                 // pool radius -> 5x5x5 kernel
constexpr float THRESH = 0.9f;

constexpr int TX       = 64;                // tile width  == blockDim.x
constexpr int TY       = 8;                 // tile height == blockDim.y
constexpr int DCHUNK   = 16;                // output planes per block
constexpr int TROWS    = TY + 2 * RAD;      // 12 rows (with halo)
constexpr int TCOLS    = TX + 2 * RAD;      // 68 cols (with halo)
constexpr int PITCH    = 72;                // padded LDS row pitch (floats)
constexpr int NTHREADS = TX * TY;           // 512 threads = 16 wave32s
constexpr int NLOAD    = TROWS * TCOLS;     // 816 elements per plane tile
constexpr int NWRAP    = NLOAD - NTHREADS;  // 304: second-slot coverage
constexpr int ZITERS   = DCHUNK + 2 * RAD;  // 20 planes streamed per block
}

__global__ __launch_bounds__(NTHREADS)
void peak3d_kernel(const float* __restrict__ in, float* __restrict__ out) {
  // double-buffered halo tiles for software-pipelined async copies
  __shared__ float tile[2][TROWS * PITCH];

  const int tx  = threadIdx.x;
  const int ty  = threadIdx.y;
  const int tid = ty * TX + tx;
  const int W0  = blockIdx.x * TX;
  const int H0  = blockIdx.y * TY;
  const int D0  = blockIdx.z * DCHUNK;

  // Every lane handles exactly TWO tile elements per plane (slot1 wraps so all
  // waves issue exactly 2 async ops -> s_wait_asynccnt arithmetic is uniform).
  const int idx0 = tid;                     // 0..511
  const int idx1 = NTHREADS + (tid % NWRAP);// 512..815 (duplicates are benign)

  const int r0 = idx0 / TCOLS, c0 = idx0 - r0 * TCOLS;
  const int r1 = idx1 / TCOLS, c1 = idx1 - r1 * TCOLS;

  // replicate-clamp (== -inf padding for stride-1 max pool) in H and W
  const int g0 = min(max(H0 + r0 - RAD, 0), Hy - 1) * Wx +
                 min(max(W0 + c0 - RAD, 0), Wx - 1);
  const int g1 = min(max(H0 + r1 - RAD, 0), Hy - 1) * Wx +
                 min(max(W0 + c1 - RAD, 0), Wx - 1);

  const int o0 = r0 * PITCH + c0;
  const int o1 = r1 * PITCH + c1;
  // LDS byte address == low 32 bits of the flat pointer (aperture rule)
  const uint32_t l0[2] = {(uint32_t)(uintptr_t)&tile[0][o0],
                          (uint32_t)(uintptr_t)&tile[1][o0]};
  const uint32_t l1[2] = {(uint32_t)(uintptr_t)&tile[0][o1],
                          (uint32_t)(uintptr_t)&tile[1][o1]};

  auto issue = [&](int z, int buf) {
    const int zc = min(max(z, 0), Dz - 1);  // replicate-clamp in depth, too
    const float* plane = in + (size_t)zc * (Hy * Wx);
#if USE_ASYNC_LDS
    asm volatile("global_load_async_to_lds_b32 %0, %1, off"
                 :: "v"(l0[buf]), "v"(plane + g0) : "memory");
    asm volatile("global_load_async_to_lds_b32 %0, %1, off"
                 :: "v"(l1[buf]), "v"(plane + g1) : "memory");
#else
    tile[buf][o0] = plane[g0];
    tile[buf][o1] = plane[g1];
#endif
  };

  const int zfirst = D0 - RAD;
  issue(zfirst, 0);                         // prime the pipeline

  float rring[5] = {0.f, 0.f, 0.f, 0.f, 0.f}; // 5x5 in-plane maxes over z
  float cring[3] = {0.f, 0.f, 0.f};           // raw center values over z

  for (int zi = 0; zi < ZITERS; ++zi) {
    const int z = zfirst + zi;

    // all waves finished reading buf[(zi+1)&1] during iteration zi-1
    __syncthreads();

    const bool more = (zi + 1 < ZITERS);
    if (more) issue(z + 1, (zi + 1) & 1);   // overlap next plane's copy

#if USE_ASYNC_LDS
    // in-order async retirement: "<=2" drains plane z, leaves z+1 in flight
    if (more) asm volatile("s_wait_asynccnt 2" ::: "memory");
    else      asm volatile("s_wait_asynccnt 0" ::: "memory");
#endif
    __syncthreads();                        // plane z visible to all waves

    const float* buf = tile[zi & 1];
    const float ctr = buf[(ty + RAD) * PITCH + (tx + RAD)];

    float cm = buf[ty * PITCH + tx];        // 5x5 in-plane max (raw values;
    #pragma unroll                          //  thresh(max) == max(thresh))
    for (int r = 0; r < 5; ++r) {
      const float* row = &buf[(ty + r) * PITCH + tx];
      float m = fmaxf(fmaxf(fmaxf(row[0], row[1]), fmaxf(row[2], row[3])), row[4]);
      cm = (r == 0) ? m : fmaxf(cm, m);
    }

    // shift depth rings; newest plane at the tail
    rring[0] = rring[1]; rring[1] = rring[2]; rring[2] = rring[3];
    rring[3] = rring[4]; rring[4] = cm;
    cring[0] = cring[1]; cring[1] = cring[2]; cring[2] = ctr;

    const int d = z - RAD;                  // output plane covered by ring
    if (d >= D0) {                          // first emit at zi==4: rings full
      float mp = fmaxf(fmaxf(fmaxf(rring[0], rring[1]),
                             fmaxf(rring[2], rring[3])), rring[4]);
      float c  = cring[0];                  // raw center of plane d
      // strict-peak + threshold collapses to a raw-value test:
      float o  = (mp > THRESH && mp == c) ? mp : 0.0f;
      out[((size_t)d * Hy + (H0 + ty)) * Wx + (W0 + tx)] = o;
    }
  }
}

extern "C" void kernel_launch(void* const* d_in, const int* in_sizes, int n_in,
                              void* d_out, int out_size, void* d_ws, size_t ws_size,
                              hipStream_t stream) {
  const float* in  = (const float*)d_in[0];   // [1,1,64,512,512] f32
  float*       out = (float*)d_out;           // [64,512,512] f32

  dim3 block(TX, TY, 1);
  dim3 grid(Wx / TX, Hy / TY, Dz / DCHUNK);   // 8 x 64 x 4 = 2048 blocks
  hipLaunchKernelGGL(peak3d_kernel, grid, block, 0, stream, in, out);
}